// PPS_loss_15358803050922
// MI455X (gfx1250) — compile-verified
//
#include <hip/hip_runtime.h>

#define H_CONST 480
#define W_CONST 640
#define MU_F 1.2f
#define NTHR 256
#define TILE 16
#define HALO (TILE + 2)          // 18
#define HALO_N (HALO * HALO)     // 324
#define TWO_PI_3 2.0943951023931953f

typedef __attribute__((ext_vector_type(2))) float v2f;
typedef __attribute__((ext_vector_type(8))) float v8f;

#ifdef __has_builtin
#if __has_builtin(__builtin_amdgcn_global_load_async_to_lds_b32)
#define USE_ASYNC_LDS 1
#endif
#if __has_builtin(__builtin_amdgcn_s_wait_asynccnt)
#define HAVE_WAIT_ASYNC_BUILTIN 1
#endif
#endif

typedef __attribute__((address_space(1))) int as1_int;
typedef __attribute__((address_space(3))) int as3_int;

// Stage one f32 from global to LDS. On gfx1250 use the async-to-LDS path
// (GLOBAL_LOAD_ASYNC_TO_LDS_B32, tracked by ASYNCcnt); fallback: plain copy.
__device__ __forceinline__ void stage_f32(const float* __restrict__ g, float* l) {
#if USE_ASYNC_LDS
    __builtin_amdgcn_global_load_async_to_lds_b32(
        (as1_int*)(void*)g, (as3_int*)l, /*imm_offset=*/0, /*cpol=*/0);
#else
    *l = *g;
#endif
}

__device__ __forceinline__ void stage_wait() {
#if USE_ASYNC_LDS
#if HAVE_WAIT_ASYNC_BUILTIN
    __builtin_amdgcn_s_wait_asynccnt(0);
#else
    asm volatile("s_wait_asynccnt 0" ::: "memory");
#endif
#endif
}

// ---------------------------------------------------------------------------
// Wave32 sum via V_WMMA_F32_16X16X4_F32:
//   A(16x4): a={s,0} -> A[m][0]=s_m, A[m][2]=s_{m+16}; B(4x16)=ones
//   => D[m][n] = s_m + s_{m+16} (independent of n). Per-lane sum of the 8
//   D VGPRs gives rows 0..7 (lanes<16) / rows 8..15 (lanes>=16); one
//   shfl_xor(16) completes the full-precision f32 wave sum.
// EXEC must be all-ones here (grid exactly tiles the image; no divergence).
// ---------------------------------------------------------------------------
__device__ __forceinline__ float wave_sum_wmma(float s) {
    v2f a; a[0] = s;    a[1] = 0.0f;
    v2f b; b[0] = 1.0f; b[1] = 1.0f;
    v8f c = {};
    c = __builtin_amdgcn_wmma_f32_16x16x4_f32(
            false, a, false, b, (short)0, c, false, false);
    float t = 0.0f;
#pragma unroll
    for (int i = 0; i < 8; ++i) t += c[i];
    t += __shfl_xor(t, 16, 32);
    return t;
}

// Smallest eigenvector of symmetric 3x3 [[a00,a01,a02],[a01,a11,a12],[a02,a12,a22]]
__device__ __forceinline__ void smallest_eigvec3(
    float a00, float a01, float a02, float a11, float a12, float a22,
    float& nx, float& ny, float& nz)
{
    float p1 = a01 * a01 + a02 * a02 + a12 * a12;
    float q  = (a00 + a11 + a22) * (1.0f / 3.0f);
    float b00 = a00 - q, b11 = a11 - q, b22 = a22 - q;
    float p2 = b00 * b00 + b11 * b11 + b22 * b22 + 2.0f * p1;
    float p  = sqrtf(p2 * (1.0f / 6.0f)) + 1e-30f;
    float ip = 1.0f / p;
    float c00 = b00 * ip, c01 = a01 * ip, c02 = a02 * ip;
    float c11 = b11 * ip, c12 = a12 * ip, c22 = b22 * ip;
    float detB = c00 * (c11 * c22 - c12 * c12)
               - c01 * (c01 * c22 - c12 * c02)
               + c02 * (c01 * c12 - c11 * c02);
    float r = fminf(1.0f, fmaxf(-1.0f, 0.5f * detB));
    float phi = acosf(r) * (1.0f / 3.0f);
    float lam = q + 2.0f * p * cosf(phi + TWO_PI_3);   // smallest eigenvalue

    float r0x = a00 - lam, r0y = a01,       r0z = a02;
    float r1x = a01,       r1y = a11 - lam, r1z = a12;
    float r2x = a02,       r2y = a12,       r2z = a22 - lam;

    float v0x = r0y * r1z - r0z * r1y, v0y = r0z * r1x - r0x * r1z, v0z = r0x * r1y - r0y * r1x;
    float v1x = r0y * r2z - r0z * r2y, v1y = r0z * r2x - r0x * r2z, v1z = r0x * r2y - r0y * r2x;
    float v2x = r1y * r2z - r1z * r2y, v2y = r1z * r2x - r1x * r2z, v2z = r1x * r2y - r1y * r2x;

    float n0 = v0x * v0x + v0y * v0y + v0z * v0z;
    float n1 = v1x * v1x + v1y * v1y + v1z * v1z;
    float n2 = v2x * v2x + v2y * v2y + v2z * v2z;

    float vx = v0x, vy = v0y, vz = v0z, nn = n0;
    if (n1 > nn) { vx = v1x; vy = v1y; vz = v1z; nn = n1; }
    if (n2 > nn) { vx = v2x; vy = v2y; vz = v2z; nn = n2; }

    float inv = rsqrtf(fmaxf(nn, 1e-38f));
    nx = vx * inv; ny = vy * inv; nz = vz * inv;

    // Deterministic sign convention (eigh sign is implementation-defined)
    float ax = fabsf(nx), ay = fabsf(ny), az = fabsf(nz);
    float dom = (ax >= ay && ax >= az) ? nx : ((ay >= az) ? ny : nz);
    if (dom < 0.0f) { nx = -nx; ny = -ny; nz = -nz; }
}

// pps at pixel (x,y) using the staged HALOxHALO LDS tile
__device__ __forceinline__ float pps_from_tile(
    const float* __restrict__ t, int lx, int ly, int x, int y,
    float inv_fx, float inv_fy, float cx, float cy)
{
    float px[9], py[9], pz[9];
    int k = 0;
#pragma unroll
    for (int dy = -1; dy <= 1; ++dy) {
        int yy = y + dy; yy = yy < 0 ? 0 : (yy > H_CONST - 1 ? H_CONST - 1 : yy);
        float v = ((float)yy - cy) * inv_fy;
#pragma unroll
        for (int dx = -1; dx <= 1; ++dx) {
            int xx = x + dx; xx = xx < 0 ? 0 : (xx > W_CONST - 1 ? W_CONST - 1 : xx);
            float u = ((float)xx - cx) * inv_fx;
            float z = t[(ly + 1 + dy) * HALO + (lx + 1 + dx)];
            px[k] = u * z; py[k] = v * z; pz[k] = z;
            ++k;
        }
    }

    float mx = 0.f, my = 0.f, mz = 0.f;
#pragma unroll
    for (int i = 0; i < 9; ++i) { mx += px[i]; my += py[i]; mz += pz[i]; }
    const float inv9 = 1.0f / 9.0f;
    mx *= inv9; my *= inv9; mz *= inv9;

    float a00 = 0.f, a01 = 0.f, a02 = 0.f, a11 = 0.f, a12 = 0.f, a22 = 0.f;
#pragma unroll
    for (int i = 0; i < 9; ++i) {
        float ux = px[i] - mx, uy = py[i] - my, uz = pz[i] - mz;
        a00 += ux * ux; a01 += ux * uy; a02 += ux * uz;
        a11 += uy * uy; a12 += uy * uz; a22 += uz * uz;
    }
    const float inv8 = 0.125f;
    a00 *= inv8; a01 *= inv8; a02 *= inv8; a11 *= inv8; a12 *= inv8; a22 *= inv8;

    float nxv, nyv, nzv;
    smallest_eigvec3(a00, a01, a02, a11, a12, a22, nxv, nyv, nzv);

    float Pz = pz[4];
    float s  = 1.0f - 1.0f / Pz;               // > 0 since z in [2,10]
    float dx_ = px[4] * s, dy_ = py[4] * s, dz_ = Pz * s;
    float q2  = dx_ * dx_ + dy_ * dy_ + dz_ * dz_;
    float idn = rsqrtf(fmaxf(q2, 1e-30f));     // 1/||d||
    float Lz  = dz_ * idn;                     // > 0
    float A   = exp2f(MU_F * log2f(Lz)) * idn; // Lz^mu / ||d||
    float dot = (dx_ * nxv + dy_ * nyv + dz_ * nzv) * idn;
    return A * dot;
}

__global__ __launch_bounds__(NTHR)
void pps_partial_kernel(
    const float* __restrict__ depth, const float* __restrict__ depth_gt,
    const float* __restrict__ Km, float* __restrict__ partials)
{
    __shared__ float tile_p[HALO_N];
    __shared__ float tile_g[HALO_N];

    const float fx = Km[0], cx = Km[2], fy = Km[4], cy = Km[5];
    const float inv_fx = 1.0f / fx, inv_fy = 1.0f / fy;

    const int lx = threadIdx.x, ly = threadIdx.y;
    const int tid = ly * TILE + lx;
    const int tx0 = blockIdx.x * TILE, ty0 = blockIdx.y * TILE;
    const size_t base = (size_t)blockIdx.z * (H_CONST * W_CONST);

    // Stage 18x18 halo of both maps (clamped) into LDS via async-to-LDS
    for (int i = tid; i < HALO_N; i += NTHR) {
        int r = i / HALO, c = i - r * HALO;
        int gy = ty0 - 1 + r; gy = gy < 0 ? 0 : (gy > H_CONST - 1 ? H_CONST - 1 : gy);
        int gx = tx0 - 1 + c; gx = gx < 0 ? 0 : (gx > W_CONST - 1 ? W_CONST - 1 : gx);
        size_t off = base + (size_t)gy * W_CONST + gx;
        stage_f32(depth + off,    &tile_p[i]);
        stage_f32(depth_gt + off, &tile_g[i]);
    }
    stage_wait();
    __syncthreads();

    const int x = tx0 + lx, y = ty0 + ly;
    float pp = pps_from_tile(tile_p, lx, ly, x, y, inv_fx, inv_fy, cx, cy);
    float pg = pps_from_tile(tile_g, lx, ly, x, y, inv_fx, inv_fy, cx, cy);
    float local = fabsf(pp - pg);

    // Wave32 reduce via WMMA (EXEC all-ones: grid exactly tiles the image)
    float wsum = wave_sum_wmma(local);

    __shared__ float sd[NTHR / 32];
    int lane = tid & 31;
    int wave = tid >> 5;
    if (lane == 0) sd[wave] = wsum;
    __syncthreads();
    if (tid == 0) {
        float t = 0.0f;
#pragma unroll
        for (int i = 0; i < NTHR / 32; ++i) t += sd[i];
        int blk = (blockIdx.z * gridDim.y + blockIdx.y) * gridDim.x + blockIdx.x;
        partials[blk] = t;
    }
}

__global__ void reduce_mean_kernel(float* __restrict__ ws, int nparts, float invN)
{
    __shared__ float sd[NTHR];
    float t = 0.0f;
    for (int i = threadIdx.x; i < nparts; i += NTHR) t += ws[i];
    sd[threadIdx.x] = t;
    __syncthreads();
    for (int s = NTHR / 2; s > 0; s >>= 1) {
        if (threadIdx.x < s) sd[threadIdx.x] += sd[threadIdx.x + s];
        __syncthreads();
    }
    if (threadIdx.x == 0) ws[nparts] = sd[0] * invN;
}

__global__ void scale_mask_kernel(
    const float* __restrict__ mask, const float* __restrict__ meanp,
    float* __restrict__ out, int N)
{
    int idx = blockIdx.x * blockDim.x + threadIdx.x;
    float m = *meanp;
    if (idx < N) out[idx] = mask[idx] * m;
}

extern "C" void kernel_launch(void* const* d_in, const int* in_sizes, int n_in,
                              void* d_out, int out_size, void* d_ws, size_t ws_size,
                              hipStream_t stream)
{
    const float* depth    = (const float*)d_in[0];
    const float* depth_gt = (const float*)d_in[1];
    const float* mask     = (const float*)d_in[2];
    const float* Km       = (const float*)d_in[3];
    float* out = (float*)d_out;
    float* wsf = (float*)d_ws;
    const int N = in_sizes[0];                 // B*H*W
    const int B = N / (H_CONST * W_CONST);     // 4
    const int gx = W_CONST / TILE;             // 40
    const int gy = H_CONST / TILE;             // 30
    const int nparts = gx * gy * B;            // 4800

    dim3 grid(gx, gy, B), block(TILE, TILE, 1);
    pps_partial_kernel<<<grid, block, 0, stream>>>(depth, depth_gt, Km, wsf);
    reduce_mean_kernel<<<1, NTHR, 0, stream>>>(wsf, nparts, 1.0f / (float)N);
    scale_mask_kernel<<<(N + NTHR - 1) / NTHR, NTHR, 0, stream>>>(mask, wsf + nparts, out, N);
}